// ReceiverAgent_19980187861590
// MI455X (gfx1250) — compile-verified
//
#include <hip/hip_runtime.h>
#include <math.h>

#define Nn   50000
#define Ee   640000
#define FEAT 128
#define EMB  128
#define HID  512
#define Bb   32

typedef float v2f __attribute__((ext_vector_type(2)));
typedef float v8f __attribute__((ext_vector_type(8)));

// ---------------------------------------------------------------- utilities
__global__ void zero_kernel(float* __restrict__ p, int n) {
    int i = blockIdx.x * blockDim.x + threadIdx.x;
    if (i < n) p[i] = 0.0f;
}

__global__ void deg_kernel(const long long* __restrict__ ei, float* __restrict__ deg) {
    int e = blockIdx.x * blockDim.x + threadIdx.x;
    if (e < Ee) {
        int d = (int)ei[(size_t)Ee + e];
        __hip_atomic_fetch_add(&deg[d], 1.0f, __ATOMIC_RELAXED, __HIP_MEMORY_SCOPE_AGENT);
    }
}

// ------------------------------------------------- dual GEMM: T=H*Wrel, R=H*Wroot
// One workgroup (128 threads = 4 waves) per 16-row block. Each wave owns 4
// 16x16 output tiles. fp32 WMMA, K stepped by 4 (V_WMMA_F32_16X16X4_F32).
// A 16x4 fp32 layout: lanes 0-15 hold (K=k,k+1), lanes 16-31 hold (K=k+2,k+3).
// B 4x16: VGPR0 = rows k / k+2 across lane halves, VGPR1 = rows k+1 / k+3.
__global__ void gemm_dual_wmma(const float* __restrict__ Hin,
                               const float* __restrict__ Wrel,
                               const float* __restrict__ Wroot,
                               float* __restrict__ T, float* __restrict__ R) {
    const int lane = threadIdx.x & 31;
    const int wave = threadIdx.x >> 5;
    const int l    = lane & 15;
    const int half = lane >> 4;
    const int row0 = blockIdx.x * 16;

    v2f a[32];
    const float* ap = Hin + (size_t)(row0 + l) * EMB + 2 * half;
#pragma unroll
    for (int ks = 0; ks < 32; ++ks) a[ks] = *(const v2f*)(ap + ks * 4);

#pragma unroll
    for (int tt = 0; tt < 4; ++tt) {
        const int tile = wave * 4 + tt;                 // 0..7 -> Wrel, 8..15 -> Wroot
        const float* W  = (tile < 8) ? Wrel : Wroot;
        float*      Out = (tile < 8) ? T    : R;
        const int col0  = (tile & 7) * 16;

        v8f c = {0.f, 0.f, 0.f, 0.f, 0.f, 0.f, 0.f, 0.f};
#pragma unroll
        for (int ks = 0; ks < 32; ++ks) {
            v2f b;
            b.x = W[(size_t)(ks * 4 + 2 * half)     * EMB + col0 + l];
            b.y = W[(size_t)(ks * 4 + 2 * half + 1) * EMB + col0 + l];
            c = __builtin_amdgcn_wmma_f32_16x16x4_f32(false, a[ks], false, b,
                                                      (short)0, c, false, false);
        }
#pragma unroll
        for (int j = 0; j < 8; ++j)
            Out[(size_t)(row0 + j + 8 * half) * EMB + col0 + l] = c[j];
    }
}

// ------------------------------------------------- edge scatter: agg[dst] += t[src]
// one wave per edge, 4 floats per lane; fp32 HW atomics.
__global__ void scatter_kernel(const float* __restrict__ t,
                               const long long* __restrict__ ei,
                               float* __restrict__ agg) {
    const int wid  = threadIdx.x >> 5;
    const int lane = threadIdx.x & 31;
    const long long e = (long long)blockIdx.x * 8 + wid;
    if (e >= Ee) return;
    const int s = (int)ei[e];
    const int d = (int)ei[(size_t)Ee + e];
    const float* srow = t   + (size_t)s * EMB;
    float*       drow = agg + (size_t)d * EMB;
#pragma unroll
    for (int j = 0; j < 4; ++j) {
        float v = srow[lane + 32 * j];
        __hip_atomic_fetch_add(&drow[lane + 32 * j], v,
                               __ATOMIC_RELAXED, __HIP_MEMORY_SCOPE_AGENT);
    }
}

// ------------------------------------------------- h = relu(agg/deg + r + bias)
__global__ void combine_kernel(const float* __restrict__ agg, const float* __restrict__ deg,
                               const float* __restrict__ r, const float* __restrict__ bias,
                               float* __restrict__ h) {
    int i = blockIdx.x * blockDim.x + threadIdx.x;
    if (i >= Nn * EMB) return;
    int node = i >> 7;
    int c    = i & 127;
    float dg = fmaxf(deg[node], 1.0f);
    float v  = agg[i] / dg + r[i] + bias[c];
    h[i] = fmaxf(v, 0.0f);
}

// ------------------------------------------------- me = last_hidden[0] @ Wp + bp
__global__ void proj_kernel(const float* __restrict__ LH, const float* __restrict__ Wp,
                            const float* __restrict__ bp, float* __restrict__ ME) {
    int i = blockIdx.x * blockDim.x + threadIdx.x;
    if (i >= Bb * EMB) return;
    int b = i >> 7;
    int c = i & 127;
    float s = bp[c];
    for (int k = 0; k < HID; ++k) s += LH[b * HID + k] * Wp[(size_t)k * EMB + c];
    ME[i] = s;
}

// ------------------------------------------------- scores = H @ ME^T  (WMMA)
__global__ void scores_wmma(const float* __restrict__ H, const float* __restrict__ ME,
                            float* __restrict__ S) {
    const int lane = threadIdx.x & 31;
    const int wave = threadIdx.x >> 5;   // 0,1 -> score cols 0-15 / 16-31
    const int l    = lane & 15;
    const int half = lane >> 4;
    const int row0 = blockIdx.x * 16;
    const int col0 = wave * 16;

    v2f a[32];
    const float* ap = H + (size_t)(row0 + l) * EMB + 2 * half;
#pragma unroll
    for (int ks = 0; ks < 32; ++ks) a[ks] = *(const v2f*)(ap + ks * 4);

    v8f c = {0.f, 0.f, 0.f, 0.f, 0.f, 0.f, 0.f, 0.f};
    // B[k][n] = ME[n][k]: a float2 load at ME[col0+l][k+2*half] yields (VGPR0,VGPR1)
    const float* bp = ME + (size_t)(col0 + l) * EMB + 2 * half;
#pragma unroll
    for (int ks = 0; ks < 32; ++ks) {
        v2f b = *(const v2f*)(bp + ks * 4);
        c = __builtin_amdgcn_wmma_f32_16x16x4_f32(false, a[ks], false, b,
                                                  (short)0, c, false, false);
    }
#pragma unroll
    for (int j = 0; j < 8; ++j)
        S[(size_t)(row0 + j + 8 * half) * Bb + col0 + l] = c[j];
}

// ------------------------------------------------- per-column max & log-sum-exp
__global__ void colstats_kernel(const float* __restrict__ S,
                                float* __restrict__ cmax, float* __restrict__ clog) {
    const int col = blockIdx.x;
    __shared__ float red[256];
    float m = -INFINITY;
    for (int n = threadIdx.x; n < Nn; n += 256) m = fmaxf(m, S[(size_t)n * Bb + col]);
    red[threadIdx.x] = m; __syncthreads();
    for (int s = 128; s > 0; s >>= 1) {
        if (threadIdx.x < s) red[threadIdx.x] = fmaxf(red[threadIdx.x], red[threadIdx.x + s]);
        __syncthreads();
    }
    float cm = red[0]; __syncthreads();
    float sum = 0.0f;
    for (int n = threadIdx.x; n < Nn; n += 256) sum += expf(S[(size_t)n * Bb + col] - cm);
    red[threadIdx.x] = sum; __syncthreads();
    for (int s = 128; s > 0; s >>= 1) {
        if (threadIdx.x < s) red[threadIdx.x] += red[threadIdx.x + s];
        __syncthreads();
    }
    if (threadIdx.x == 0) { cmax[col] = cm; clog[col] = logf(red[0]); }
}

__global__ void finalize_kernel(float* __restrict__ S, const float* __restrict__ cmax,
                                const float* __restrict__ clog) {
    int i = blockIdx.x * blockDim.x + threadIdx.x;
    if (i >= Nn * Bb) return;
    int c = i & 31;
    S[i] = S[i] - cmax[c] - clog[c];
}

// ---------------------------------------------------------------- launch
extern "C" void kernel_launch(void* const* d_in, const int* in_sizes, int n_in,
                              void* d_out, int out_size, void* d_ws, size_t ws_size,
                              hipStream_t stream) {
    const float*     x   = (const float*)d_in[0];
    const long long* ei  = (const long long*)d_in[1];   // int64 [2, E]
    const float*     lh  = (const float*)d_in[2];       // [1, B, HID]
    const float*     W1r = (const float*)d_in[3];
    const float*     W1o = (const float*)d_in[4];
    const float*     b1  = (const float*)d_in[5];
    const float*     W2r = (const float*)d_in[6];
    const float*     W2o = (const float*)d_in[7];
    const float*     b2  = (const float*)d_in[8];
    const float*     Wp  = (const float*)d_in[9];
    const float*     bp  = (const float*)d_in[10];
    float* out = (float*)d_out;

    float* ws   = (float*)d_ws;
    float* tbuf = ws;                                    // [N, EMB]
    float* rbuf = tbuf + (size_t)Nn * EMB;               // [N, EMB]
    float* agg  = rbuf + (size_t)Nn * EMB;               // [N, EMB]
    float* deg  = agg  + (size_t)Nn * EMB;               // [N]   (contiguous after agg)
    float* hbuf = deg  + Nn;                             // [N, EMB]
    float* me   = hbuf + (size_t)Nn * EMB;               // [B, EMB]
    float* cmax = me   + Bb * EMB;                       // [B]
    float* clog = cmax + Bb;                             // [B]

    const int ZN1 = Nn * EMB + Nn;                        // agg + deg together
    const int ZN2 = Nn * EMB;

    // ---- degree (shared by both layers) + clear accumulators
    zero_kernel<<<(ZN1 + 255) / 256, 256, 0, stream>>>(agg, ZN1);
    deg_kernel<<<(Ee + 255) / 256, 256, 0, stream>>>(ei, deg);

    // ---- layer 1
    gemm_dual_wmma<<<Nn / 16, 128, 0, stream>>>(x, W1r, W1o, tbuf, rbuf);
    scatter_kernel<<<Ee / 8, 256, 0, stream>>>(tbuf, ei, agg);
    combine_kernel<<<(Nn * EMB + 255) / 256, 256, 0, stream>>>(agg, deg, rbuf, b1, hbuf);

    // ---- layer 2
    zero_kernel<<<(ZN2 + 255) / 256, 256, 0, stream>>>(agg, ZN2);
    gemm_dual_wmma<<<Nn / 16, 128, 0, stream>>>(hbuf, W2r, W2o, tbuf, rbuf);
    scatter_kernel<<<Ee / 8, 256, 0, stream>>>(tbuf, ei, agg);
    combine_kernel<<<(Nn * EMB + 255) / 256, 256, 0, stream>>>(agg, deg, rbuf, b2, hbuf);

    // ---- projection + scores + log_softmax(axis=0)
    proj_kernel<<<(Bb * EMB + 255) / 256, 256, 0, stream>>>(lh, Wp, bp, me);
    scores_wmma<<<Nn / 16, 64, 0, stream>>>(hbuf, me, out);
    colstats_kernel<<<Bb, 256, 0, stream>>>(out, cmax, clog);
    finalize_kernel<<<(Nn * Bb + 255) / 256, 256, 0, stream>>>(out, cmax, clog);
}